// dcrnn_55362128445522
// MI455X (gfx1250) — compile-verified
//
#include <hip/hip_runtime.h>
#include <hip/hip_bf16.h>
#include <math.h>

typedef __attribute__((ext_vector_type(2))) float v2f;
typedef __attribute__((ext_vector_type(8))) float v8f;

#define N_NODES 325
#define NPAD    336          // 21*16, rows/K padded for WMMA
#define BATCH   128
#define HID     64
#define SEQT    12
#define INDIM   2
#define FP0     68           // (INDIM+HID)=66 padded to mult-of-4
#define FP1     128          // (HID+HID)
#define NPB     (NPAD*BATCH) // 43008 = 2688*16

static inline int ceil_div(int a, int b) { return (a + b - 1) / b; }

// ---------------- elementwise / setup kernels ----------------

__global__ void deg_kernel(const float* __restrict__ adj, float* __restrict__ deg) {
    int t = blockIdx.x * blockDim.x + threadIdx.x;
    if (t < N_NODES) {                       // row sums
        float s = 0.f;
        const float* r = adj + (size_t)t * N_NODES;
        for (int j = 0; j < N_NODES; ++j) s += r[j];
        deg[t] = s;
    } else if (t < 2 * N_NODES) {            // col sums
        int c = t - N_NODES;
        float s = 0.f;
        for (int j = 0; j < N_NODES; ++j) s += adj[(size_t)j * N_NODES + c];
        deg[t] = s;
    }
}

// S0[m,n] = A[n,m]/rowsum(A)[n] ; S1[m,n] = A[m,n]/colsum(A)[n]; zero-padded to NPAD
__global__ void build_supports(const float* __restrict__ adj, const float* __restrict__ deg,
                               float* __restrict__ S0, float* __restrict__ S1) {
    int idx = blockIdx.x * blockDim.x + threadIdx.x;
    if (idx >= NPAD * NPAD) return;
    int m = idx / NPAD, n = idx % NPAD;
    float s0 = 0.f, s1 = 0.f;
    if (m < N_NODES && n < N_NODES) {
        float d  = deg[n];            float di  = d  > 0.f ? 1.f / d  : 0.f;
        float d2 = deg[N_NODES + n];  float d2i = d2 > 0.f ? 1.f / d2 : 0.f;
        s0 = adj[(size_t)n * N_NODES + m] * di;
        s1 = adj[(size_t)m * N_NODES + n] * d2i;
    }
    S0[idx] = s0; S1[idx] = s1;
}

// pad layer-0 weights: src [5*66, O] -> dst [5][FP0][O] (rows 66..67 zero)
__global__ void pad_w0(float* __restrict__ dst, const float* __restrict__ src, int O) {
    int total = 5 * FP0 * O;
    int idx = blockIdx.x * blockDim.x + threadIdx.x;
    if (idx >= total) return;
    int o = idx % O;
    int r = (idx / O) % FP0;
    int k = idx / (O * FP0);
    dst[idx] = (r < 66) ? src[((size_t)(k * 66 + r)) * O + o] : 0.f;
}

// h[(n*B+b)*H+o] <- hinit[b*N*H + n*H + o], zero for padded nodes
__global__ void init_h(float* __restrict__ h, const float* __restrict__ hin) {
    int idx = blockIdx.x * blockDim.x + threadIdx.x;
    if (idx >= NPB * HID) return;
    int o = idx % HID;
    int b = (idx / HID) % BATCH;
    int n = idx / (HID * BATCH);
    h[idx] = (n < N_NODES) ? hin[((size_t)b * N_NODES + n) * HID + o] : 0.f;
}

// layer-0 xh pack: [NPAD,B,FP0] = [x(2) | (r*)h(64) | pad(2)], zero padded rows
__global__ void pack0(float* __restrict__ xh, const float* __restrict__ xin,
                      const float* __restrict__ h, const float* __restrict__ ru) {
    int idx = blockIdx.x * blockDim.x + threadIdx.x;
    if (idx >= NPB * FP0) return;
    int f = idx % FP0;
    int b = (idx / FP0) % BATCH;
    int n = idx / (FP0 * BATCH);
    float v = 0.f;
    if (n < N_NODES) {
        if (f < INDIM) {
            v = xin[((size_t)b * N_NODES + n) * INDIM + f];
        } else if (f < INDIM + HID) {
            int o = f - INDIM;
            v = h[((size_t)n * BATCH + b) * HID + o];
            if (ru) v *= ru[((size_t)n * BATCH + b) * (2 * HID) + o];  // r gate
        }
    }
    xh[idx] = v;
}

// layer-1 xh pack: [NPAD,B,FP1] = [x(=h0,64) | (r*)h1(64)]
__global__ void pack1(float* __restrict__ xh, const float* __restrict__ x,
                      const float* __restrict__ h, const float* __restrict__ ru) {
    int idx = blockIdx.x * blockDim.x + threadIdx.x;
    if (idx >= NPB * FP1) return;
    int f = idx % FP1;
    int b = (idx / FP1) % BATCH;
    int n = idx / (FP1 * BATCH);
    float v = 0.f;
    if (n < N_NODES) {
        if (f < HID) {
            v = x[((size_t)n * BATCH + b) * HID + f];
        } else {
            int o = f - HID;
            v = h[((size_t)n * BATCH + b) * HID + o];
            if (ru) v *= ru[((size_t)n * BATCH + b) * (2 * HID) + o];
        }
    }
    xh[idx] = v;
}

__global__ void init_bias(float* __restrict__ out, const float* __restrict__ bias, int O) {
    int idx = blockIdx.x * blockDim.x + threadIdx.x;
    if (idx >= NPB * O) return;
    out[idx] = bias[idx % O];
}

__global__ void sigmoid_k(float* __restrict__ p, int n) {
    int idx = blockIdx.x * blockDim.x + threadIdx.x;
    if (idx < n) p[idx] = 1.f / (1.f + expf(-p[idx]));
}

// h = u*h + (1-u)*tanh(cacc)
__global__ void update_h(float* __restrict__ h, const float* __restrict__ ru,
                         const float* __restrict__ cacc) {
    int idx = blockIdx.x * blockDim.x + threadIdx.x;
    if (idx >= NPB * HID) return;
    int o  = idx % HID;
    int rb = idx / HID;                       // n*B + b
    float u = ru[(size_t)rb * (2 * HID) + HID + o];
    float c = tanhf(cacc[idx]);
    h[idx] = u * h[idx] + (1.f - u) * c;
}

// dst[b*N*H + n*H + o] = h[(n*B+b)*H + o]
__global__ void write_bnh(float* __restrict__ dst, const float* __restrict__ h) {
    int idx = blockIdx.x * blockDim.x + threadIdx.x;
    if (idx >= BATCH * N_NODES * HID) return;
    int o = idx % HID;
    int n = (idx / HID) % N_NODES;
    int b = idx / (HID * N_NODES);
    dst[idx] = h[((size_t)n * BATCH + b) * HID + o];
}

// ---------------- WMMA f32 GEMM (software double-buffered) ----------------
// One wave per 16x64 C tile (4 v8f accumulators = 32 VGPRs). All dims exactly
// tiled (M%16==0, Ncols%64==0, K%4==0, K>=8) so EXEC stays all-ones.
// The next k-step's A (1x b64) and B (8x b32) fragments are prefetched into a
// fresh register set before the 4 chained WMMAs consume the current set, so
// load latency overlaps the matrix ops instead of serializing them.
// MODE 0: C = A*B ; MODE 1: C = 2*A*B - D ; MODE 2: C += A*B
template <int MODE>
__global__ __launch_bounds__(32) void wmma_gemm(const float* __restrict__ A, int lda,
                                                const float* __restrict__ B, int ldb,
                                                float* __restrict__ C, int ldc,
                                                const float* __restrict__ D, int K) {
    const int lane = threadIdx.x;
    const int ml   = lane & 15;
    const int half = lane >> 4;          // 0 or 1
    const int m0   = blockIdx.y << 4;
    const int n0   = blockIdx.x << 6;

    // A frag layout (16x4 f32): lane L -> M=L&15 ; VGPR0/1 = K {2*half, 2*half+1}
    const float* Ap = A + (size_t)(m0 + ml) * lda + 2 * half;
    // B frag layout (4x16 f32): lane L -> N=L&15 ; VGPR0/1 = K {2*half, 2*half+1}
    const float* Bp = B + (size_t)(2 * half) * ldb + n0 + ml;

    v8f acc0 = {}, acc1 = {}, acc2 = {}, acc3 = {};

    // ---- preload k = 0 ----
    v2f a0 = *(const v2f*)(Ap);               // 8B-aligned: lda even
    v2f b00, b01, b02, b03;
    b00.x = Bp[0];  b00.y = Bp[ldb + 0];
    b01.x = Bp[16]; b01.y = Bp[ldb + 16];
    b02.x = Bp[32]; b02.y = Bp[ldb + 32];
    b03.x = Bp[48]; b03.y = Bp[ldb + 48];

    for (int k = 4; k < K; k += 4) {
        // prefetch next k-step into distinct registers
        v2f a1 = *(const v2f*)(Ap + k);
        const float* bp = Bp + (size_t)k * ldb;
        v2f b10, b11, b12, b13;
        b10.x = bp[0];  b10.y = bp[ldb + 0];
        b11.x = bp[16]; b11.y = bp[ldb + 16];
        b12.x = bp[32]; b12.y = bp[ldb + 32];
        b13.x = bp[48]; b13.y = bp[ldb + 48];
        // compute current step while prefetch is in flight
        acc0 = __builtin_amdgcn_wmma_f32_16x16x4_f32(false, a0, false, b00, (short)0, acc0, false, false);
        acc1 = __builtin_amdgcn_wmma_f32_16x16x4_f32(false, a0, false, b01, (short)0, acc1, false, false);
        acc2 = __builtin_amdgcn_wmma_f32_16x16x4_f32(false, a0, false, b02, (short)0, acc2, false, false);
        acc3 = __builtin_amdgcn_wmma_f32_16x16x4_f32(false, a0, false, b03, (short)0, acc3, false, false);
        a0 = a1; b00 = b10; b01 = b11; b02 = b12; b03 = b13;
    }
    // ---- last k-step ----
    acc0 = __builtin_amdgcn_wmma_f32_16x16x4_f32(false, a0, false, b00, (short)0, acc0, false, false);
    acc1 = __builtin_amdgcn_wmma_f32_16x16x4_f32(false, a0, false, b01, (short)0, acc1, false, false);
    acc2 = __builtin_amdgcn_wmma_f32_16x16x4_f32(false, a0, false, b02, (short)0, acc2, false, false);
    acc3 = __builtin_amdgcn_wmma_f32_16x16x4_f32(false, a0, false, b03, (short)0, acc3, false, false);

#pragma unroll
    for (int j = 0; j < 4; ++j) {
        v8f acc = (j == 0) ? acc0 : (j == 1) ? acc1 : (j == 2) ? acc2 : acc3;
#pragma unroll
        for (int v = 0; v < 8; ++v) {
            // C/D frag: VGPR v -> row v + 8*half ; col = lane&15
            size_t off = (size_t)(m0 + v + 8 * half) * ldc + (n0 + j * 16 + ml);
            float val = acc[v];
            if (MODE == 1) val = 2.f * val - D[off];
            else if (MODE == 2) val += C[off];
            C[off] = val;
        }
    }
}

// ---------------- host orchestration ----------------

// One diffusion-convolution: out[NPB,O] = bias + sum_k term_k @ Wblk_k
// terms: xh, S0 xh, 2 S0(S0 xh)-xh, S1 xh, 2 S1(S1 xh)-xh
static void launch_dconv(hipStream_t stream, const float* S0, const float* S1,
                         float* xh, float* t1, float* t2,
                         const float* W, size_t wblk, int ldw, const float* bias,
                         float* out, int O, int FP) {
    const int BF = BATCH * FP;                 // diffusion column count (mult of 64)
    dim3 blk(32);
    dim3 gdiff(BF / 64, NPAD / 16);
    dim3 gproj(O / 64, NPB / 16);
    const int ebs = 256;
    init_bias<<<ceil_div(NPB * O, ebs), ebs, 0, stream>>>(out, bias, O);
    wmma_gemm<2><<<gproj, blk, 0, stream>>>(xh, FP, W + 0 * wblk, ldw, out, O, nullptr, FP);
    wmma_gemm<0><<<gdiff, blk, 0, stream>>>(S0, NPAD, xh, BF, t1, BF, nullptr, NPAD);
    wmma_gemm<2><<<gproj, blk, 0, stream>>>(t1, FP, W + 1 * wblk, ldw, out, O, nullptr, FP);
    wmma_gemm<1><<<gdiff, blk, 0, stream>>>(S0, NPAD, t1, BF, t2, BF, xh, NPAD);
    wmma_gemm<2><<<gproj, blk, 0, stream>>>(t2, FP, W + 2 * wblk, ldw, out, O, nullptr, FP);
    wmma_gemm<0><<<gdiff, blk, 0, stream>>>(S1, NPAD, xh, BF, t1, BF, nullptr, NPAD);
    wmma_gemm<2><<<gproj, blk, 0, stream>>>(t1, FP, W + 3 * wblk, ldw, out, O, nullptr, FP);
    wmma_gemm<1><<<gdiff, blk, 0, stream>>>(S1, NPAD, t1, BF, t2, BF, xh, NPAD);
    wmma_gemm<2><<<gproj, blk, 0, stream>>>(t2, FP, W + 4 * wblk, ldw, out, O, nullptr, FP);
}

extern "C" void kernel_launch(void* const* d_in, const int* in_sizes, int n_in,
                              void* d_out, int out_size, void* d_ws, size_t ws_size,
                              hipStream_t stream) {
    const float* inputs = (const float*)d_in[0];
    const float* hinit  = (const float*)d_in[1];
    const float* adj    = (const float*)d_in[2];
    const float* Wg0    = (const float*)d_in[3];
    const float* bg0    = (const float*)d_in[4];
    const float* Wc0    = (const float*)d_in[5];
    const float* bc0    = (const float*)d_in[6];
    const float* Wg1    = (const float*)d_in[7];
    const float* bg1    = (const float*)d_in[8];
    const float* Wc1    = (const float*)d_in[9];
    const float* bc1    = (const float*)d_in[10];
    float* out = (float*)d_out;

    float* ws = (float*)d_ws;
    size_t off = 0;
    auto alloc = [&](size_t n) { float* p = ws + off; off += n; return p; };
    float* S0   = alloc((size_t)NPAD * NPAD);
    float* S1   = alloc((size_t)NPAD * NPAD);
    float* deg  = alloc(2 * N_NODES);
    float* Wg0p = alloc((size_t)5 * FP0 * 2 * HID);
    float* Wc0p = alloc((size_t)5 * FP0 * HID);
    float* xh   = alloc((size_t)NPB * FP1);
    float* t1   = alloc((size_t)NPB * FP1);
    float* t2   = alloc((size_t)NPB * FP1);
    float* ru   = alloc((size_t)NPB * 2 * HID);
    float* cacc = alloc((size_t)NPB * HID);
    float* h0   = alloc((size_t)NPB * HID);
    float* h1   = alloc((size_t)NPB * HID);
    if (off * sizeof(float) > ws_size) return;  // workspace too small

    const int ebs = 256;
    deg_kernel<<<ceil_div(2 * N_NODES, ebs), ebs, 0, stream>>>(adj, deg);
    build_supports<<<ceil_div(NPAD * NPAD, ebs), ebs, 0, stream>>>(adj, deg, S0, S1);
    pad_w0<<<ceil_div(5 * FP0 * 2 * HID, ebs), ebs, 0, stream>>>(Wg0p, Wg0, 2 * HID);
    pad_w0<<<ceil_div(5 * FP0 * HID, ebs), ebs, 0, stream>>>(Wc0p, Wc0, HID);
    init_h<<<ceil_div(NPB * HID, ebs), ebs, 0, stream>>>(h0, hinit);
    init_h<<<ceil_div(NPB * HID, ebs), ebs, 0, stream>>>(h1, hinit + (size_t)BATCH * N_NODES * HID);

    const size_t seq_base = (size_t)2 * BATCH * N_NODES * HID;  // after out_hidden [L,B,N*H]
    for (int t = 0; t < SEQT; ++t) {
        const float* xin = inputs + (size_t)t * BATCH * N_NODES * INDIM;
        // ---- layer 0 ----
        pack0<<<ceil_div(NPB * FP0, ebs), ebs, 0, stream>>>(xh, xin, h0, nullptr);
        launch_dconv(stream, S0, S1, xh, t1, t2, Wg0p, (size_t)FP0 * 2 * HID, 2 * HID, bg0,
                     ru, 2 * HID, FP0);
        sigmoid_k<<<ceil_div(NPB * 2 * HID, ebs), ebs, 0, stream>>>(ru, NPB * 2 * HID);
        pack0<<<ceil_div(NPB * FP0, ebs), ebs, 0, stream>>>(xh, xin, h0, ru);
        launch_dconv(stream, S0, S1, xh, t1, t2, Wc0p, (size_t)FP0 * HID, HID, bc0,
                     cacc, HID, FP0);
        update_h<<<ceil_div(NPB * HID, ebs), ebs, 0, stream>>>(h0, ru, cacc);
        // ---- layer 1 (input = new h0) ----
        pack1<<<ceil_div(NPB * FP1, ebs), ebs, 0, stream>>>(xh, h0, h1, nullptr);
        launch_dconv(stream, S0, S1, xh, t1, t2, Wg1, (size_t)FP1 * 2 * HID, 2 * HID, bg1,
                     ru, 2 * HID, FP1);
        sigmoid_k<<<ceil_div(NPB * 2 * HID, ebs), ebs, 0, stream>>>(ru, NPB * 2 * HID);
        pack1<<<ceil_div(NPB * FP1, ebs), ebs, 0, stream>>>(xh, h0, h1, ru);
        launch_dconv(stream, S0, S1, xh, t1, t2, Wc1, (size_t)FP1 * HID, HID, bc1,
                     cacc, HID, FP1);
        update_h<<<ceil_div(NPB * HID, ebs), ebs, 0, stream>>>(h1, ru, cacc);
        // out_seq[t] = h1
        write_bnh<<<ceil_div(BATCH * N_NODES * HID, ebs), ebs, 0, stream>>>(
            out + seq_base + (size_t)t * BATCH * N_NODES * HID, h1);
    }
    // out_hidden[0]=h0, out_hidden[1]=h1
    write_bnh<<<ceil_div(BATCH * N_NODES * HID, ebs), ebs, 0, stream>>>(out, h0);
    write_bnh<<<ceil_div(BATCH * N_NODES * HID, ebs), ebs, 0, stream>>>(
        out + (size_t)BATCH * N_NODES * HID, h1);
}